// LlamaAttention_15925738733622
// MI455X (gfx1250) — compile-verified
//
#include <hip/hip_runtime.h>
#include <stdint.h>

// ---------------- problem constants (match reference) ----------------
#define B_      32
#define HID_    4096
#define H_      32
#define KVH_    8
#define D_      128
#define G_      4            // H / KVH
#define BS_     16
#define NB_     256
#define QKVN_   ((H_ + 2 * KVH_) * D_)   // 6144
#define SCALE_  0.08838834764831845f     // D^-0.5
#define THETA_  10000.0f
#define NEG_INF_ (-1.0e30f)

typedef float v2f __attribute__((ext_vector_type(2)));
typedef float v8f __attribute__((ext_vector_type(8)));

// =====================================================================
// WMMA f32 16x16x4 GEMM tile: C[16x16] += A[16xK] * B[KxN] tile
// A layout (ISA 7.12.2, 32-bit A 16x4): lanes 0-15 hold M=lane, K=0/1 in
// v0/v1; lanes 16-31 hold M=lane-16, K=2/3.  B (4x16): lanes 0-15 hold
// rows K=0/1 at N=lane; lanes 16-31 rows K=2/3.  C/D: VGPR r -> M = r +
// 8*(lane>=16), N = lane&15.
// =====================================================================
__device__ inline void gemm_tile_wmma_f32(const float* __restrict__ A, int lda,
                                          const float* __restrict__ Bm, int ldb,
                                          float* __restrict__ C, int ldc,
                                          int K, int m0, int n0) {
  const int lane  = threadIdx.x & 31;
  const int idx16 = lane & 15;
  const int khalf = (lane >> 4) << 1;   // 0 or 2
  const float* arow = A + (size_t)(m0 + idx16) * lda + khalf;
  const float* bcol = Bm + (size_t)khalf * ldb + n0 + idx16;
  v8f acc = {};
  for (int k = 0; k < K; k += 4) {
    v2f a, b;
    a.x = arow[k];
    a.y = arow[k + 1];
    b.x = bcol[(size_t)k * ldb];
    b.y = bcol[(size_t)k * ldb + ldb];
    if ((k & 127) == 0 && (k + 128) < K)
      __builtin_prefetch(bcol + (size_t)(k + 128) * ldb, 0, 1);
    acc = __builtin_amdgcn_wmma_f32_16x16x4_f32(
        /*neg_a=*/false, a, /*neg_b=*/false, b,
        /*c_mod=*/(short)0, acc, /*reuse_a=*/false, /*reuse_b=*/false);
  }
  const int mhalf = (lane >> 4) << 3;   // 0 or 8
  float* crow = C + (size_t)(m0 + mhalf) * ldc + n0 + idx16;
#pragma unroll
  for (int r = 0; r < 8; ++r) crow[(size_t)r * ldc] = acc[r];
}

// qkv = hidden[32,4096] @ Wqkv[4096,6144]
__global__ __launch_bounds__(128) void qkv_gemm_kernel(
    const float* __restrict__ hidden, const float* __restrict__ Wqkv,
    float* __restrict__ qkv) {
  const int wave = threadIdx.x >> 5;
  const int n0 = blockIdx.x * 64 + wave * 16;
  const int m0 = blockIdx.y * 16;
  gemm_tile_wmma_f32(hidden, HID_, Wqkv, QKVN_, qkv, QKVN_, HID_, m0, n0);
}

// out = attn[32,4096] @ Wo[4096,4096]
__global__ __launch_bounds__(128) void o_gemm_kernel(
    const float* __restrict__ attn, const float* __restrict__ Wo,
    float* __restrict__ out) {
  const int wave = threadIdx.x >> 5;
  const int n0 = blockIdx.x * 64 + wave * 16;
  const int m0 = blockIdx.y * 16;
  gemm_tile_wmma_f32(attn, H_ * D_, Wo, HID_, out, HID_, H_ * D_, m0, n0);
}

// =====================================================================
// Neox-style RoPE on q (heads 0..31) and k (heads 32..39) of qkv, in place
// =====================================================================
__global__ void rope_kernel(float* __restrict__ qkv,
                            const int* __restrict__ positions) {
  const int b = blockIdx.x;
  const float pos = (float)positions[b];
  float* base = qkv + (size_t)b * QKVN_;
  for (int i = threadIdx.x; i < (H_ + KVH_) * (D_ / 2); i += blockDim.x) {
    const int head = i >> 6;   // / (D/2)
    const int dd   = i & 63;
    const float inv = __powf(THETA_, -2.0f * (float)dd / (float)D_);
    float s, c;
    __sincosf(pos * inv, &s, &c);
    float* p = base + head * D_;
    const float x1 = p[dd];
    const float x2 = p[dd + 64];
    p[dd]      = x1 * c - x2 * s;
    p[dd + 64] = x2 * c + x1 * s;
  }
}

// =====================================================================
// Flash-decode paged attention. One block per (kv_head, batch); 8 waves
// split 16-token chunks.  Each wave stages its K chunk into a private
// LDS buffer via CDNA5 async global->LDS copies (ASYNCcnt), then runs
// online softmax out of LDS; V accumulation uses direct coalesced loads.
// =====================================================================
#define AT_WAVES 8
#define KROW_    132                    // 128 + 4 pad: distinct banks per token
#define KCHUNK_  (16 * KROW_)           // floats per wave stage buffer

__device__ inline float dot16(const float4 k0, const float4 k1,
                              const float4 k2, const float4 k3,
                              const float* __restrict__ q) {
  return k0.x * q[0]  + k0.y * q[1]  + k0.z * q[2]  + k0.w * q[3]
       + k1.x * q[4]  + k1.y * q[5]  + k1.z * q[6]  + k1.w * q[7]
       + k2.x * q[8]  + k2.y * q[9]  + k2.z * q[10] + k2.w * q[11]
       + k3.x * q[12] + k3.y * q[13] + k3.z * q[14] + k3.w * q[15];
}

__global__ __launch_bounds__(256) void paged_attn_kernel(
    const float* __restrict__ qkv, const float* __restrict__ k_cache,
    const float* __restrict__ v_cache, const int* __restrict__ block_tables,
    const int* __restrict__ context_lens, float* __restrict__ attn_out) {
  const int kvh  = blockIdx.x;
  const int b    = blockIdx.y;
  const int tid  = threadIdx.x;
  const int wave = tid >> 5;
  const int lane = tid & 31;
  const int j    = lane & 15;   // token within chunk
  const int half = lane >> 4;   // which 64-dim half of D this lane dots

  __shared__ __align__(16) float q_s[G_ * D_];
  __shared__ __align__(16) float k_stage[AT_WAVES * KCHUNK_];   // ~66 KB of 320 KB WGP LDS
  __shared__ float part_m[AT_WAVES * G_];
  __shared__ float part_l[AT_WAVES * G_];
  __shared__ __align__(16) float part_o[AT_WAVES * G_ * D_];

  const float* qkv_b = qkv + (size_t)b * QKVN_;
  for (int i = tid; i < G_ * D_; i += blockDim.x)
    q_s[i] = qkv_b[(kvh * G_) * D_ + i];
  __syncthreads();

  const float* knew = qkv_b + H_ * D_ + kvh * D_;
  const float* vnew = qkv_b + (H_ + KVH_) * D_ + kvh * D_;

  const int ctx = context_lens[b];
  const int L   = ctx + 1;          // cached tokens + the new token

  // LDS byte offset of this wave's staging buffer (low 32 bits of the
  // generic pointer to LDS == LDS byte address on CDNA5).
  float* kbuf = &k_stage[wave * KCHUNK_];
  const uint32_t kbuf_base = (uint32_t)(uintptr_t)kbuf;

  float m[G_], l[G_], o[G_][4];
#pragma unroll
  for (int g = 0; g < G_; ++g) {
    m[g] = NEG_INF_; l[g] = 0.f;
    o[g][0] = o[g][1] = o[g][2] = o[g][3] = 0.f;
  }

  const int nchunk = (L + 15) >> 4;
  for (int c = wave; c < nchunk; c += AT_WAVES) {
    const int t0 = c << 4;

    // ---- stage K[t0..t0+15][0..127] -> LDS, fully coalesced async copies.
    // Previous chunk's ds reads must land before the async copies rewrite the buffer.
    asm volatile("s_wait_dscnt 0x0" ::: "memory");
#pragma unroll
    for (int i = 0; i < 16; ++i) {
      int tt = t0 + i;
      tt = tt > ctx ? ctx : tt;            // clamp tail: valid addr, masked later
      const float* kp;
      if (tt == ctx) {
        kp = knew;                         // new token's roped K (workspace)
      } else {
        const int blk = block_tables[b * NB_ + (tt >> 4)];
        kp = k_cache + (((size_t)blk * BS_ + (tt & 15)) * KVH_ + kvh) * D_;
      }
      const float*   gsrc = kp + lane * 4;                       // 16B per lane
      const uint32_t ldst = kbuf_base + (uint32_t)((i * KROW_ + lane * 4) * 4);
      asm volatile("global_load_async_to_lds_b128 %0, %1, off"
                   :: "v"(ldst), "v"(gsrc) : "memory");
    }
    asm volatile("s_wait_asynccnt 0x0" ::: "memory");

    // ---- scores from LDS: lane (j, half) dots 64 dims of token t0+j
    const int  t     = t0 + j;
    const bool valid = t < L;
    float sc[G_] = {0.f, 0.f, 0.f, 0.f};
    {
      const float* kh = kbuf + j * KROW_ + half * 64;
#pragma unroll
      for (int d0 = 0; d0 < 64; d0 += 16) {
        const float4 k0 = *(const float4*)(kh + d0);
        const float4 k1 = *(const float4*)(kh + d0 + 4);
        const float4 k2 = *(const float4*)(kh + d0 + 8);
        const float4 k3 = *(const float4*)(kh + d0 + 12);
#pragma unroll
        for (int g = 0; g < G_; ++g)
          sc[g] += dot16(k0, k1, k2, k3, q_s + g * D_ + half * 64 + d0);
      }
    }
    // combine the two D-halves (lanes j and j+16 hold the halves)
#pragma unroll
    for (int g = 0; g < G_; ++g) {
      sc[g] += __shfl_xor(sc[g], 16);
      sc[g] = valid ? sc[g] * SCALE_ : NEG_INF_;
    }
    // online softmax update per head
    float pk[G_];
#pragma unroll
    for (int g = 0; g < G_; ++g) {
      float x = sc[g];
      x = fmaxf(x, __shfl_xor(x, 1));
      x = fmaxf(x, __shfl_xor(x, 2));
      x = fmaxf(x, __shfl_xor(x, 4));
      x = fmaxf(x, __shfl_xor(x, 8));
      const float mnew = fmaxf(m[g], x);
      const float p = valid ? __expf(sc[g] - mnew) : 0.f;
      float ps = p;
      ps += __shfl_xor(ps, 1);
      ps += __shfl_xor(ps, 2);
      ps += __shfl_xor(ps, 4);
      ps += __shfl_xor(ps, 8);
      const float co = __expf(m[g] - mnew);
      l[g] = l[g] * co + ps;
      m[g] = mnew;
      o[g][0] *= co; o[g][1] *= co; o[g][2] *= co; o[g][3] *= co;
      pk[g] = p;
    }
    // accumulate P*V; V loads fully coalesced (v[t][lane + 32u])
    const int tmax = min(16, L - t0);
    for (int jj = 0; jj < tmax; ++jj) {
      const int t2 = t0 + jj;
      const float* vp;
      if (t2 == ctx) {
        vp = vnew;
      } else {
        const int blk = block_tables[b * NB_ + (t2 >> 4)];
        vp = v_cache + (((size_t)blk * BS_ + (t2 & 15)) * KVH_ + kvh) * D_;
      }
      const float pj0 = __shfl(pk[0], jj);
      const float pj1 = __shfl(pk[1], jj);
      const float pj2 = __shfl(pk[2], jj);
      const float pj3 = __shfl(pk[3], jj);
#pragma unroll
      for (int u = 0; u < 4; ++u) {
        const float vv = vp[u * 32 + lane];
        o[0][u] += pj0 * vv;
        o[1][u] += pj1 * vv;
        o[2][u] += pj2 * vv;
        o[3][u] += pj3 * vv;
      }
    }
  }

  // publish per-wave partials
  if (lane == 0) {
#pragma unroll
    for (int g = 0; g < G_; ++g) {
      part_m[wave * G_ + g] = m[g];
      part_l[wave * G_ + g] = l[g];
    }
  }
#pragma unroll
  for (int g = 0; g < G_; ++g)
#pragma unroll
    for (int u = 0; u < 4; ++u)
      part_o[(wave * G_ + g) * D_ + u * 32 + lane] = o[g][u];
  __syncthreads();

  // merge 8 wave partials -> final attention output
  for (int i = tid; i < G_ * D_; i += blockDim.x) {
    const int g = i >> 7;        // / D_
    const int d = i & (D_ - 1);
    float M = NEG_INF_;
#pragma unroll
    for (int w = 0; w < AT_WAVES; ++w) M = fmaxf(M, part_m[w * G_ + g]);
    float Lt = 0.f, Od = 0.f;
#pragma unroll
    for (int w = 0; w < AT_WAVES; ++w) {
      const float e = __expf(part_m[w * G_ + g] - M);
      Lt += e * part_l[w * G_ + g];
      Od += e * part_o[(w * G_ + g) * D_ + d];
    }
    attn_out[((size_t)b * H_ + kvh * G_ + g) * D_ + d] = Od / Lt;
  }
}

// =====================================================================
extern "C" void kernel_launch(void* const* d_in, const int* in_sizes, int n_in,
                              void* d_out, int out_size, void* d_ws, size_t ws_size,
                              hipStream_t stream) {
  const float* hidden       = (const float*)d_in[0];
  const int*   positions    = (const int*)d_in[1];
  const float* k_cache      = (const float*)d_in[2];
  const float* v_cache      = (const float*)d_in[3];
  const int*   block_tables = (const int*)d_in[4];
  const int*   context_lens = (const int*)d_in[5];
  const float* Wqkv         = (const float*)d_in[6];
  const float* Wo           = (const float*)d_in[7];
  float* out = (float*)d_out;

  float* qkv  = (float*)d_ws;                       // [B, QKVN]
  float* attn = qkv + (size_t)B_ * QKVN_;           // [B, H*D]

  qkv_gemm_kernel<<<dim3(QKVN_ / 64, B_ / 16), 128, 0, stream>>>(hidden, Wqkv, qkv);
  rope_kernel<<<B_, 256, 0, stream>>>(qkv, positions);
  paged_attn_kernel<<<dim3(KVH_, B_), 256, 0, stream>>>(
      qkv, k_cache, v_cache, block_tables, context_lens, attn);
  o_gemm_kernel<<<dim3(HID_ / 64, B_ / 16), 128, 0, stream>>>(attn, Wo, out);
}